// GQA_88708254531831
// MI455X (gfx1250) — compile-verified
//
#include <hip/hip_runtime.h>
#include <math.h>

typedef __attribute__((ext_vector_type(2))) float v2f;
typedef __attribute__((ext_vector_type(8))) float v8f;

#define HIDDEN  1024
#define SEQ     2048
#define BATCH   2
#define NGROUPS 16
#define HDIM    64

__device__ __forceinline__ v8f wmma_f32(v2f a, v2f b, v8f c) {
    // D = A(16x4 f32) * B(4x16 f32) + C(16x16 f32)
    return __builtin_amdgcn_wmma_f32_16x16x4_f32(
        /*neg_a=*/false, a, /*neg_b=*/false, b,
        /*c_mod=*/(short)0, c, /*reuse_a=*/false, /*reuse_b=*/false);
}

// Async global->LDS 16B copy (GVS mode): saddr = wave-uniform 64-bit base,
// vaddr = per-lane 32-bit byte offset, first operand = per-lane LDS byte addr.
// Tracked by ASYNCcnt; publish with s_wait_asynccnt 0 before the barrier.
__device__ __forceinline__ void async_load_b128(void* ldsPtr, const void* gBase,
                                                unsigned byteOff) {
    unsigned ldsOff = (unsigned)(size_t)ldsPtr;   // low 32 bits = LDS offset
    asm volatile("global_load_async_to_lds_b128 %0, %1, %2"
                 :
                 : "v"(ldsOff), "v"(byteOff),
                   "s"((unsigned long long)(size_t)gBase)
                 : "memory");
}

__device__ __forceinline__ void wait_async0() {
    asm volatile("s_wait_asynccnt 0" ::: "memory");
}

// ---------------------------------------------------------------------------
// C[M,N] = A[M,K] @ W[N,K]^T + bias[N]
// Block tile: 128(M) x 64(N), 256 threads = 8 waves, K staged in LDS by 16
// via async global->LDS copies. Wave w computes rows [w*16, w*16+16) x 64.
// ---------------------------------------------------------------------------
__global__ __launch_bounds__(256) void gemm_bias_kernel(
    const float* __restrict__ A, const float* __restrict__ W,
    const float* __restrict__ bias, float* __restrict__ C,
    int M, int N, int K)
{
    __shared__ float As[128 * 20];   // 128x16, stride 20 (16B-aligned rows)
    __shared__ float Ws[64 * 20];    // 64x16

    const int tid  = threadIdx.x;
    const int wave = tid >> 5;
    const int lane = tid & 31;
    const int hi   = lane >> 4;      // lane half
    const int ln   = lane & 15;
    const int mBase = blockIdx.y * 128;
    const int nBase = blockIdx.x * 64;

    v8f acc[4] = {};

    for (int k0 = 0; k0 < K; k0 += 16) {
        __syncthreads();             // previous iteration's LDS reads done
        const float* aBase = A + (size_t)mBase * K + k0;
        const float* wBase = W + (size_t)nBase * K + k0;
        // A tile: 128 rows x 4 float4 = 512 vecs / 256 threads
        #pragma unroll
        for (int i = 0; i < 2; ++i) {
            int vid = tid + i * 256;
            int row = vid >> 2, c4 = vid & 3;
            async_load_b128(&As[row * 20 + c4 * 4], aBase,
                            (unsigned)((row * K + c4 * 4) * 4));
        }
        // W tile: 64 rows x 4 float4 = 256 vecs
        {
            int row = tid >> 2, c4 = tid & 3;
            async_load_b128(&Ws[row * 20 + c4 * 4], wBase,
                            (unsigned)((row * K + c4 * 4) * 4));
        }
        wait_async0();
        __syncthreads();

        #pragma unroll
        for (int ks = 0; ks < 4; ++ks) {
            const int kk = ks * 4 + 2 * hi;      // A/B fragment: K = vgpr + 2*half
            v2f a;
            a.x = As[(wave * 16 + ln) * 20 + kk];
            a.y = As[(wave * 16 + ln) * 20 + kk + 1];
            #pragma unroll
            for (int nt = 0; nt < 4; ++nt) {
                v2f b;                            // B[k][n] = W[n][k]
                b.x = Ws[(nt * 16 + ln) * 20 + kk];
                b.y = Ws[(nt * 16 + ln) * 20 + kk + 1];
                acc[nt] = wmma_f32(a, b, acc[nt]);
            }
        }
    }

    // C/D layout: row = vgpr + 8*half, col = lane&15
    #pragma unroll
    for (int nt = 0; nt < 4; ++nt) {
        #pragma unroll
        for (int v = 0; v < 8; ++v) {
            int row = mBase + wave * 16 + v + 8 * hi;
            int col = nBase + nt * 16 + ln;
            C[(size_t)row * N + col] = acc[nt][v] + bias[col];
        }
    }
}

// ---------------------------------------------------------------------------
// In-place RoPE over the full hidden dim (half = 512), one thread per pair.
// ---------------------------------------------------------------------------
__global__ __launch_bounds__(256) void rope_kernel(float* __restrict__ buf, int totalRows)
{
    const int half = HIDDEN / 2;
    int t = blockIdx.x * blockDim.x + threadIdx.x;
    if (t >= totalRows * half) return;
    int row = t / half;
    int i   = t - row * half;
    int pos = row & (SEQ - 1);
    float inv_freq = __powf(10000.0f, -(float)i / (float)half);
    float ang = (float)pos * inv_freq;
    float s, c;
    __sincosf(ang, &s, &c);
    float* p = buf + (size_t)row * HIDDEN;
    float x1 = p[i], x2 = p[i + half];
    p[i]        = x1 * c - x2 * s;
    p[i + half] = x2 * c + x1 * s;
}

// ---------------------------------------------------------------------------
// Flash attention: grid = (SEQ/128, BATCH*NGROUPS), 256 threads = 8 waves.
// Wave w owns query rows [blockIdx.x*128 + w*16, +16). Q fragments resident in
// registers. 64-key K/V tiles staged in LDS via async copies; P re-laid out
// through a per-wave LDS region for the PV WMMA.
// ---------------------------------------------------------------------------
__global__ __launch_bounds__(256) void attn_kernel(
    const float* __restrict__ q, const float* __restrict__ k,
    const float* __restrict__ v, const int* __restrict__ mask,
    float* __restrict__ o)
{
    __shared__ float Ks[64 * 68];        // 64 keys x 64 dims (stride 68)
    __shared__ float Vs[64 * 68];
    __shared__ float Ps[8 * 16 * 68];    // per-wave 16x64 P tile

    const int tid  = threadIdx.x;
    const int wave = tid >> 5;
    const int lane = tid & 31;
    const int hi   = lane >> 4;
    const int ln   = lane & 15;
    const int bb   = blockIdx.y >> 4;    // batch
    const int gg   = blockIdx.y & 15;    // head/group
    const int qbase = blockIdx.x * 128 + wave * 16;
    const float scale = 0.125f;          // 1/sqrt(64)

    const size_t headOff = (size_t)gg * HDIM;
    const size_t batRow  = (size_t)bb * SEQ;

    // Load Q fragments: A layout, M = ln, K = 4*t2 + vgpr + 2*hi
    v2f aq[16];
    {
        const float* qr = q + (batRow + qbase + ln) * HIDDEN + headOff;
        #pragma unroll
        for (int t2 = 0; t2 < 16; ++t2) {
            aq[t2].x = qr[t2 * 4 + 2 * hi];
            aq[t2].y = qr[t2 * 4 + 2 * hi + 1];
        }
    }

    v8f oacc[4] = {};
    float mrow[8], lrow[8];
    #pragma unroll
    for (int vv = 0; vv < 8; ++vv) { mrow[vv] = -1e30f; lrow[vv] = 0.0f; }

    float* Pw = &Ps[wave * 16 * 68];

    for (int kt = 0; kt < SEQ / 64; ++kt) {
        __syncthreads();
        // Stage K and V tiles (64 rows x 16 float4 each) via async copies
        const float* kBase = k + (batRow + kt * 64) * HIDDEN + headOff;
        const float* vBase = v + (batRow + kt * 64) * HIDDEN + headOff;
        #pragma unroll
        for (int i = 0; i < 4; ++i) {
            int vid = tid + i * 256;
            int row = vid >> 4, c4 = vid & 15;
            unsigned goff = (unsigned)((row * HIDDEN + c4 * 4) * 4);
            async_load_b128(&Ks[row * 68 + c4 * 4], kBase, goff);
            async_load_b128(&Vs[row * 68 + c4 * 4], vBase, goff);
        }
        wait_async0();
        __syncthreads();

        // S = Q @ K^T : 4 tiles of 16x16, contraction over head_dim (16 steps of 4)
        v8f sacc[4] = {};
        #pragma unroll
        for (int t2 = 0; t2 < 16; ++t2) {
            const int kk = t2 * 4 + 2 * hi;
            #pragma unroll
            for (int nt = 0; nt < 4; ++nt) {
                v2f b;                           // B[k][n] = Ktile[key=n][dim=k]
                b.x = Ks[(nt * 16 + ln) * 68 + kk];
                b.y = Ks[(nt * 16 + ln) * 68 + kk + 1];
                sacc[nt] = wmma_f32(aq[t2], b, sacc[nt]);
            }
        }

        // Mask + scale (key mask; column = lane&15 within each n-tile)
        float p[4][8];
        #pragma unroll
        for (int nt = 0; nt < 4; ++nt) {
            int mk = mask[bb * SEQ + kt * 64 + nt * 16 + ln];
            #pragma unroll
            for (int vv = 0; vv < 8; ++vv)
                p[nt][vv] = mk ? sacc[nt][vv] * scale : -1e30f;
        }

        // Online softmax; row = vv + 8*hi lives across 16 lanes of one half
        float corr[8];
        #pragma unroll
        for (int vv = 0; vv < 8; ++vv) {
            float tmax = fmaxf(fmaxf(p[0][vv], p[1][vv]), fmaxf(p[2][vv], p[3][vv]));
            #pragma unroll
            for (int off = 1; off < 16; off <<= 1)
                tmax = fmaxf(tmax, __shfl_xor(tmax, off, 32));
            float mnew = fmaxf(mrow[vv], tmax);
            corr[vv] = __expf(mrow[vv] - mnew);
            mrow[vv] = mnew;
            float rsum = 0.0f;
            #pragma unroll
            for (int nt = 0; nt < 4; ++nt) {
                p[nt][vv] = __expf(p[nt][vv] - mnew);
                rsum += p[nt][vv];
            }
            #pragma unroll
            for (int off = 1; off < 16; off <<= 1)
                rsum += __shfl_xor(rsum, off, 32);
            lrow[vv] = lrow[vv] * corr[vv] + rsum;
        }
        #pragma unroll
        for (int nt = 0; nt < 4; ++nt)
            #pragma unroll
            for (int vv = 0; vv < 8; ++vv)
                oacc[nt][vv] *= corr[vv];

        // Spill P (C layout) to per-wave LDS as row-major 16x64
        #pragma unroll
        for (int nt = 0; nt < 4; ++nt)
            #pragma unroll
            for (int vv = 0; vv < 8; ++vv)
                Pw[(vv + 8 * hi) * 68 + nt * 16 + ln] = p[nt][vv];
        __syncthreads();

        // O += P @ V : contraction over the 64 keys (16 steps of 4)
        #pragma unroll
        for (int t2 = 0; t2 < 16; ++t2) {
            const int kk = t2 * 4 + 2 * hi;
            v2f ap;                              // A layout from P
            ap.x = Pw[ln * 68 + kk];
            ap.y = Pw[ln * 68 + kk + 1];
            #pragma unroll
            for (int dt = 0; dt < 4; ++dt) {
                v2f bv;                          // B[k][n] = Vtile[key=k][dim=n]
                bv.x = Vs[kk * 68 + dt * 16 + ln];
                bv.y = Vs[(kk + 1) * 68 + dt * 16 + ln];
                oacc[dt] = wmma_f32(ap, bv, oacc[dt]);
            }
        }
    }

    // Normalize and store: o laid out (b, s, hidden) with head-contiguous dims
    #pragma unroll
    for (int vv = 0; vv < 8; ++vv) lrow[vv] = 1.0f / lrow[vv];
    #pragma unroll
    for (int dt = 0; dt < 4; ++dt) {
        #pragma unroll
        for (int vv = 0; vv < 8; ++vv) {
            int row = qbase + vv + 8 * hi;
            size_t col = headOff + dt * 16 + ln;
            o[(batRow + row) * HIDDEN + col] = oacc[dt][vv] * lrow[vv];
        }
    }
}

// ---------------------------------------------------------------------------
extern "C" void kernel_launch(void* const* d_in, const int* in_sizes, int n_in,
                              void* d_out, int out_size, void* d_ws, size_t ws_size,
                              hipStream_t stream)
{
    const float* x    = (const float*)d_in[0];
    const int*   amsk = (const int*)  d_in[1];
    const float* Wq   = (const float*)d_in[2];
    const float* bq   = (const float*)d_in[3];
    const float* Wk   = (const float*)d_in[4];
    const float* bk   = (const float*)d_in[5];
    const float* Wv   = (const float*)d_in[6];
    const float* bv   = (const float*)d_in[7];
    const float* Wo   = (const float*)d_in[8];
    const float* bo   = (const float*)d_in[9];
    float* out = (float*)d_out;

    const int Mtok = BATCH * SEQ;                 // 4096 tokens
    const size_t bufElems = (size_t)Mtok * HIDDEN;
    float* qb = (float*)d_ws;
    float* kb = qb + bufElems;
    float* vb = kb + bufElems;
    float* ob = vb + bufElems;

    dim3 gGrid(HIDDEN / 64, Mtok / 128);          // (16, 32)

    gemm_bias_kernel<<<gGrid, 256, 0, stream>>>(x, Wq, bq, qb, Mtok, HIDDEN, HIDDEN);
    gemm_bias_kernel<<<gGrid, 256, 0, stream>>>(x, Wk, bk, kb, Mtok, HIDDEN, HIDDEN);
    gemm_bias_kernel<<<gGrid, 256, 0, stream>>>(x, Wv, bv, vb, Mtok, HIDDEN, HIDDEN);

    int ropeWork = Mtok * (HIDDEN / 2);
    rope_kernel<<<(ropeWork + 255) / 256, 256, 0, stream>>>(qb, Mtok);
    rope_kernel<<<(ropeWork + 255) / 256, 256, 0, stream>>>(kb, Mtok);

    attn_kernel<<<dim3(SEQ / 128, BATCH * NGROUPS), 256, 0, stream>>>(qb, kb, vb, amsk, ob);

    gemm_bias_kernel<<<gGrid, 256, 0, stream>>>(ob, Wo, bo, out, Mtok, HIDDEN, HIDDEN);
}